// StockGraph_65859028517059
// MI455X (gfx1250) — compile-verified
//
#include <hip/hip_runtime.h>
#include <hip/hip_bf16.h>

typedef float v2f __attribute__((ext_vector_type(2)));
typedef float v8f __attribute__((ext_vector_type(8)));

#define KEY_NEG_INF 0x007FFFFFu  // monotone key of -inf

__device__ __forceinline__ unsigned int fkey(float f) {
    unsigned int b = __float_as_uint(f);
    return b ^ ((b & 0x80000000u) ? 0xFFFFFFFFu : 0x80000000u);
}
__device__ __forceinline__ float fdecode(unsigned int k) {
    unsigned int b = (k & 0x80000000u) ? (k ^ 0x80000000u) : ~k;
    return __uint_as_float(b);
}

// ---------------------------------------------------------------------------
// Init: out = 0, denom = 0, emax_key = key(-inf)
// ---------------------------------------------------------------------------
__global__ __launch_bounds__(256) void init_kernel(float* __restrict__ out,
                                                   unsigned int* __restrict__ emax_key,
                                                   float* __restrict__ denom,
                                                   int n_out, int n_nodes) {
    int i = blockIdx.x * blockDim.x + threadIdx.x;
    if (i < n_out) out[i] = 0.0f;
    if (i < n_nodes) {
        emax_key[i] = KEY_NEG_INF;
        denom[i] = 0.0f;
    }
}

// ---------------------------------------------------------------------------
// C[N x 128] = A[N x K] * B[K x 128] using V_WMMA_F32_16X16X4_F32.
// One wave32 per 16x16 output tile; 8 waves/block cover 128 output columns
// of a 16-row stripe. A-frag per ISA 7.12.2: lanes 0-15 carry K={0,1},
// lanes 16-31 carry K={2,3}; B/C/D are row-striped across lanes.
// ---------------------------------------------------------------------------
template <int K>
__global__ __launch_bounds__(256) void gemm_wmma_kernel(const float* __restrict__ A,
                                                        const float* __restrict__ B,
                                                        float* __restrict__ C) {
    const int lane = threadIdx.x & 31;
    const int wave = threadIdx.x >> 5;
    const int r0 = blockIdx.x * 16;     // output row tile
    const int c0 = wave * 16;           // output col tile (8 waves x 16 = 128)
    const int m  = lane & 15;           // row (A) / col (B,C,D) within tile
    const int hi = lane >> 4;           // 0 -> K pair {0,1}, 1 -> K pair {2,3}

    v8f acc = {};
    const float* arow = A + (size_t)(r0 + m) * K;
#pragma unroll
    for (int k0 = 0; k0 < K; k0 += 4) {
        v2f a = *(const v2f*)(arow + k0 + 2 * hi);      // 8B-aligned pair
        v2f b;
        b.x = B[(size_t)(k0 + 2 * hi    ) * 128 + c0 + m];
        b.y = B[(size_t)(k0 + 2 * hi + 1) * 128 + c0 + m];
        acc = __builtin_amdgcn_wmma_f32_16x16x4_f32(
            /*neg_a=*/false, a, /*neg_b=*/false, b,
            /*c_mod=*/(short)0, acc, /*reuse_a=*/false, /*reuse_b=*/false);
    }
#pragma unroll
    for (int j = 0; j < 8; ++j) {
        C[(size_t)(r0 + j + 8 * hi) * 128 + c0 + m] = acc[j];
    }
}

// ---------------------------------------------------------------------------
// One wave32 per edge: e = dot(z[src], dz[dst]) over 128 dims, then
// segment-max via order-preserving uint atomicMax.
// ---------------------------------------------------------------------------
__global__ __launch_bounds__(256) void edge_dot_kernel(const float* __restrict__ z,
                                                       const float* __restrict__ dz,
                                                       const int* __restrict__ src,
                                                       const int* __restrict__ dst,
                                                       float* __restrict__ e,
                                                       unsigned int* __restrict__ emax_key,
                                                       int E) {
    const int lane = threadIdx.x & 31;
    const int w = blockIdx.x * 8 + (threadIdx.x >> 5);
    if (w >= E) return;
    const int s = src[w], d = dst[w];
    float4 a = ((const float4*)(z  + (size_t)s * 128))[lane];
    float4 b = ((const float4*)(dz + (size_t)d * 128))[lane];
    float t = a.x * b.x + a.y * b.y + a.z * b.z + a.w * b.w;
#pragma unroll
    for (int off = 16; off > 0; off >>= 1)
        t += __shfl_xor(t, off, 32);
    if (lane == 0) {
        e[w] = t;
        atomicMax(&emax_key[d], fkey(t));
    }
}

// ---------------------------------------------------------------------------
// Per edge: ex = exp(e - emax[dst]) (in place), denom[dst] += ex.
// Nodes with no in-edges keep the init key -> emax treated as 0 (matches
// the reference's isfinite fixup).
// ---------------------------------------------------------------------------
__global__ __launch_bounds__(256) void edge_exp_kernel(const int* __restrict__ dst,
                                                       const unsigned int* __restrict__ emax_key,
                                                       float* __restrict__ e_inout,
                                                       float* __restrict__ denom,
                                                       int E) {
    int i = blockIdx.x * blockDim.x + threadIdx.x;
    if (i >= E) return;
    const int d = dst[i];
    const unsigned int k = emax_key[d];
    const float emax = (k == KEY_NEG_INF) ? 0.0f : fdecode(k);
    const float ex = __expf(e_inout[i] - emax);
    e_inout[i] = ex;
    atomicAdd(&denom[d], ex);
}

// ---------------------------------------------------------------------------
// One wave32 per edge: out[dst] += (ex/denom[dst]) * z[src] (128 floats,
// 4 atomic f32 adds per lane; out fits in the 192MB L2).
// ---------------------------------------------------------------------------
__global__ __launch_bounds__(256) void edge_scatter_kernel(const float* __restrict__ z,
                                                           const float* __restrict__ ex,
                                                           const float* __restrict__ denom,
                                                           const int* __restrict__ src,
                                                           const int* __restrict__ dst,
                                                           float* __restrict__ out,
                                                           int E) {
    const int lane = threadIdx.x & 31;
    const int w = blockIdx.x * 8 + (threadIdx.x >> 5);
    if (w >= E) return;
    const int s = src[w], d = dst[w];
    const float den = denom[d];
    const float alpha = ex[w] / (den == 0.0f ? 1.0f : den);
    float4 a = ((const float4*)(z + (size_t)s * 128))[lane];
    float* o = out + (size_t)d * 128 + lane * 4;
    atomicAdd(o + 0, alpha * a.x);
    atomicAdd(o + 1, alpha * a.y);
    atomicAdd(o + 2, alpha * a.z);
    atomicAdd(o + 3, alpha * a.w);
}

// ---------------------------------------------------------------------------
// Host launcher
// ---------------------------------------------------------------------------
extern "C" void kernel_launch(void* const* d_in, const int* in_sizes, int n_in,
                              void* d_out, int out_size, void* d_ws, size_t ws_size,
                              hipStream_t stream) {
    const float* h     = (const float*)d_in[0];   // [N,128]
    const float* feat  = (const float*)d_in[1];   // [N,64]
    const float* W_fc  = (const float*)d_in[2];   // [128,128]
    const float* W_dst = (const float*)d_in[3];   // [64,128]
    const int*   src   = (const int*)d_in[4];     // [E]
    const int*   dst   = (const int*)d_in[5];     // [E]
    float* out = (float*)d_out;

    const int N = in_sizes[0] / 128;
    const int E = in_sizes[4];

    // Workspace layout (4-byte aligned throughout)
    char* ws = (char*)d_ws;
    float* z        = (float*)ws;                              // N*128
    float* dz       = (float*)(ws + (size_t)N * 128 * 4);      // N*128
    float* e        = (float*)(ws + (size_t)2 * N * 128 * 4);  // E (reused as ex)
    unsigned int* emax_key = (unsigned int*)(ws + (size_t)2 * N * 128 * 4 + (size_t)E * 4);
    float* denom    = (float*)((char*)emax_key + (size_t)N * 4);

    const int n_out = N * 128;

    // 1. init accumulators
    {
        int total = n_out;  // covers N too (N <= N*128)
        int blocks = (total + 255) / 256;
        init_kernel<<<blocks, 256, 0, stream>>>(out, emax_key, denom, n_out, N);
    }
    // 2. z = h @ W_fc   (K=128)
    gemm_wmma_kernel<128><<<N / 16, 256, 0, stream>>>(h, W_fc, z);
    // 3. dz = feat @ W_dst (K=64)
    gemm_wmma_kernel<64><<<N / 16, 256, 0, stream>>>(feat, W_dst, dz);
    // 4. per-edge dot + segment max
    edge_dot_kernel<<<(E + 7) / 8, 256, 0, stream>>>(z, dz, src, dst, e, emax_key, E);
    // 5. exp + denom
    edge_exp_kernel<<<(E + 255) / 256, 256, 0, stream>>>(dst, emax_key, e, denom, E);
    // 6. weighted scatter into out
    edge_scatter_kernel<<<(E + 7) / 8, 256, 0, stream>>>(z, e, denom, src, dst, out, E);
}